// AFNO1D_2954937499964
// MI455X (gfx1250) — compile-verified
//
#include <hip/hip_runtime.h>
#include <math.h>

// ---------------------------------------------------------------------------
// AFNO1D for MI455X (gfx1250, wave32).
//   B=4, L=2048, P=512, C=8, NUM_BLOCKS=8, BLOCK_SIZE=64, Lf=1025
// Pipeline: rfft (Stockham, LDS) -> block-diag complex MLP (fp32 WMMA, TDM
// staging) -> irfft + residual.  Bandwidth-bound (~660 MB @ 23.3 TB/s), so
// fp32 WMMA (full accuracy) costs nothing vs low-precision paths.
// ---------------------------------------------------------------------------

#define B_   4
#define L_   2048
#define P_   512
#define C_   8
#define NB_  8
#define BS_  64
#define LF_  1025               // L/2 + 1
#define ROWS_ (B_ * LF_ * C_)   // 32800 rows per block (= 16 * 2050 exactly)

typedef float v2f __attribute__((ext_vector_type(2)));
typedef float v8f __attribute__((ext_vector_type(8)));
typedef unsigned int u32x4 __attribute__((ext_vector_type(4)));
typedef int i32x4 __attribute__((ext_vector_type(4)));
typedef int i32x8 __attribute__((ext_vector_type(8)));

#define ORTHO_SCALE 0.02209708691207961f   // 1/sqrt(2048)

#if defined(__has_builtin)
#if __has_builtin(__builtin_amdgcn_tensor_load_to_lds) && \
    __has_builtin(__builtin_amdgcn_s_wait_tensorcnt)
#define HAVE_TDM 1
#endif
#endif
#ifndef HAVE_TDM
#define HAVE_TDM 0
#endif

#if HAVE_TDM
// ---------------------------------------------------------------------------
// Tensor Data Mover: DMA a 2-D fp32 tile (tile_w x tile_h, global row stride
// stride_elems) into LDS.  pad1 inserts 1 DWORD of LDS padding per 64 DWORDs
// (pad_interval code 5, pad_amount code 0) -> rows land on a 65-float pitch,
// matching our bank-conflict-free WMMA fragment layout.
// D# packing per CDNA5 ISA ch.8 (group0 128b, group1 256b; groups 2/3 unused
// for <=2-D tensors).  Issued wave-uniform; tracked by TENSORcnt.
// ---------------------------------------------------------------------------
__device__ static inline void tdm_load_2d(const float* gsrc, float* ldst,
                                          unsigned tile_w, unsigned tile_h,
                                          unsigned stride_elems, bool pad1)
{
    // group 0: count=1 | lds_addr | global_addr[56:0] | type=2
    u32x4 g0;
    unsigned long long ga = (unsigned long long)gsrc;
    g0[0] = 1u;                                          // count=1, user desc
    g0[1] = (unsigned)(unsigned long long)ldst;          // low 32b == LDS offset
    g0[2] = (unsigned)ga;
    g0[3] = (unsigned)((ga >> 32) & 0x01FFFFFFu) | (2u << 30);  // type=2 (image)

    // group 1
    unsigned w0 = (2u << 16);                            // data_size = 4 bytes
    if (pad1) w0 |= (1u << 20) | (5u << 22);             // pad_en, 64 dw, +1 dw
    const unsigned td0 = stride_elems;                   // tensor dim0 (OOB)
    const unsigned td1 = tile_h;                         // tensor dim1
    i32x8 g1;
    g1[0] = (int)w0;
    g1[1] = (int)((td0 & 0xFFFFu) << 16);                // tensor_dim0 [79:48]
    g1[2] = (int)((td0 >> 16) | ((td1 & 0xFFFFu) << 16));// dim0 hi | dim1 lo
    g1[3] = (int)((td1 >> 16) | (tile_w << 16));         // dim1 hi | tile_dim0
    g1[4] = (int)(tile_h > 1 ? tile_h : 0);              // tile_dim1 (0=unused)
    g1[5] = (int)stride_elems;                           // tensor_dim0_stride
    g1[6] = 0;
    g1[7] = 0;

    i32x4 z4 = {0, 0, 0, 0};
#if __clang_major__ >= 23
    i32x8 z8 = {0, 0, 0, 0, 0, 0, 0, 0};
    __builtin_amdgcn_tensor_load_to_lds(g0, g1, z4, z4, z8, 0);
#else
    __builtin_amdgcn_tensor_load_to_lds(g0, g1, z4, z4, 0);
#endif
}
#endif // HAVE_TDM

// ---------------------------------------------------------------------------
// Kernel 1: forward real FFT (complex FFT of real input), ortho norm.
// One workgroup per (b,p,c) signal. 256 threads, radix-2 Stockham in LDS.
// Output layout: ws[( (b*LF_ + l)*C_ + c )*P_ + p]  (rows contiguous in p).
// ---------------------------------------------------------------------------
__global__ __launch_bounds__(256)
void fft_fwd_kernel(const float* __restrict__ x,
                    float* __restrict__ gr, float* __restrict__ gi)
{
    __shared__ float sre[2][2048];
    __shared__ float sim[2][2048];

    const int wg  = blockIdx.x;          // 0 .. B*P*C-1
    const int b   = wg >> 12;            // / (P_*C_)
    const int rem = wg & 4095;
    const int p   = rem >> 3;
    const int c   = rem & 7;
    const int tid = threadIdx.x;

    const float* base = x + (size_t)b * (L_ * P_ * C_) + (size_t)p * C_ + c;

    // prefetch the strided column, then load (stride P_*C_ = 4096 floats)
    #pragma unroll
    for (int i = 0; i < 8; ++i) {
        int l = tid + (i << 8);
        __builtin_prefetch(base + (size_t)l * (P_ * C_), 0, 0);
    }
    #pragma unroll
    for (int i = 0; i < 8; ++i) {
        int l = tid + (i << 8);
        sre[0][l] = base[(size_t)l * (P_ * C_)];
        sim[0][l] = 0.0f;
    }
    __syncthreads();

    // Stockham radix-2, 11 stages, sign = -1 (forward)
    int cur = 0;
    int logm = 0;
    #pragma unroll 1
    for (int stage = 0; stage < 11; ++stage) {
        const int m = 1 << logm;
        #pragma unroll
        for (int it = 0; it < 4; ++it) {
            const int t = tid + (it << 8);          // 0..1023 butterflies
            const int j = t >> logm;
            const int k = t & (m - 1);
            const float ang = -3.14159265358979323846f *
                              ((float)(j * m) * (1.0f / 1024.0f));
            const float wr = __cosf(ang);
            const float wi = __sinf(ang);
            const int i0 = k + (j << logm);
            const float ar = sre[cur][i0],        ai = sim[cur][i0];
            const float br = sre[cur][i0 + 1024], bi = sim[cur][i0 + 1024];
            const int o0 = k + (j << (logm + 1));
            sre[cur ^ 1][o0] = ar + br;
            sim[cur ^ 1][o0] = ai + bi;
            const float dr = ar - br, di = ai - bi;
            sre[cur ^ 1][o0 + m] = wr * dr - wi * di;
            sim[cur ^ 1][o0 + m] = wr * di + wi * dr;
        }
        __syncthreads();
        cur ^= 1;
        ++logm;
    }

    // store bins 0..1024 with ortho scaling
    #pragma unroll
    for (int i = 0; i < 5; ++i) {
        int l = tid + (i << 8);
        if (l < LF_) {
            size_t o = ((size_t)((b * LF_ + l) * C_ + c)) * P_ + p;
            gr[o] = sre[cur][l] * ORTHO_SCALE;
            gi[o] = sim[cur][l] * ORTHO_SCALE;
        }
    }
}

// ---------------------------------------------------------------------------
// fp32 WMMA GEMM pass over one 16(M) x 16(N) tile, K=64 in steps of 4.
// A: LDS rows padded to 65 floats. W: LDS 64x64 row-major [k][n].
// ISA layout (wave32): A frag = 2 VGPR/lane, lanes 0-15 hold K={kb,kb+1},
// lanes 16-31 hold K={kb+2,kb+3}; B frag symmetric; C/D = 8 VGPR/lane.
// bsign folds the complex-mul subtraction into the B load (f32 WMMA has no
// A/B negate modifier, only CNeg).
// ---------------------------------------------------------------------------
__device__ static inline v8f wmma_tile(const float* A, const float* W,
                                       float bsign, v8f acc,
                                       int lcol, int half, int ncol0)
{
    #pragma unroll
    for (int kb = 0; kb < 64; kb += 4) {
        const int k0 = kb + 2 * half;
        v2f a, bb;
        a.x  = A[lcol * 65 + k0];
        a.y  = A[lcol * 65 + k0 + 1];
        bb.x = bsign * W[k0 * 64 + ncol0 + lcol];
        bb.y = bsign * W[(k0 + 1) * 64 + ncol0 + lcol];
        acc = __builtin_amdgcn_wmma_f32_16x16x4_f32(
                  false, a, false, bb, (short)0, acc, false, false);
    }
    return acc;
}

// ---------------------------------------------------------------------------
// Kernel 2: block-diagonal complex 2-layer MLP on the spectrum (in place).
// grid = (2050 row-tiles, 8 blocks), 128 threads = 4 waves, each wave one
// 16x16 N-subtile of the 16x64 output tile.  A tiles and weights staged via
// the Tensor Data Mover (wave 0 issues, TENSORcnt + barrier publishes).
// ---------------------------------------------------------------------------
__global__ __launch_bounds__(128)
void afno_mlp_kernel(float* __restrict__ gr, float* __restrict__ gi,
                     const float* __restrict__ w1, const float* __restrict__ b1,
                     const float* __restrict__ w2, const float* __restrict__ b2)
{
    __shared__ __align__(16) float Wr[4096];   // current-layer real weights 64x64
    __shared__ __align__(16) float Wi[4096];   // current-layer imag weights 64x64
    __shared__ __align__(16) float Ar[16 * 65];
    __shared__ __align__(16) float Ai[16 * 65];
    __shared__ float Hr[16 * 65];
    __shared__ float Hi[16 * 65];

    const int tid = threadIdx.x;
    const int n   = blockIdx.y;              // block 0..7
    const int r0  = blockIdx.x * 16;         // row-tile base (rows exact: 16*2050)
    const size_t abase = (size_t)r0 * P_ + (size_t)n * BS_;

#if HAVE_TDM
    // ---- TDM staging: A tiles (2-D, padded to 65-float pitch) + layer-1 weights
    if (tid < 32) {
        tdm_load_2d(gr + abase, Ar, BS_, 16, P_, true);
        tdm_load_2d(gi + abase, Ai, BS_, 16, P_, true);
        tdm_load_2d(w1 + (size_t)n * 4096,         Wr, 4096, 1, 4096, false);
        tdm_load_2d(w1 + (size_t)(NB_ + n) * 4096, Wi, 4096, 1, 4096, false);
        __builtin_amdgcn_s_wait_tensorcnt(0);
    }
#else
    // ---- fallback: per-lane staging
    #pragma unroll
    for (int j = 0; j < 8; ++j) {
        int idx  = tid + (j << 7);
        int row  = idx >> 6;
        int colk = idx & 63;
        size_t goff = abase + (size_t)row * P_ + colk;
        Ar[row * 65 + colk] = gr[goff];
        Ai[row * 65 + colk] = gi[goff];
    }
    {
        const float4* w1r4 = (const float4*)(w1 + (size_t)n * 4096);
        const float4* w1i4 = (const float4*)(w1 + (size_t)(NB_ + n) * 4096);
        float4* Wr4 = (float4*)Wr;
        float4* Wi4 = (float4*)Wi;
        #pragma unroll
        for (int j = 0; j < 8; ++j) {
            int idx = tid + (j << 7);
            Wr4[idx] = w1r4[idx];
            Wi4[idx] = w1i4[idx];
        }
    }
#endif
    __syncthreads();

    const int lane = tid & 31;
    const int ni   = tid >> 5;               // wave id -> N-subtile 0..3
    const int half = lane >> 4;
    const int lcol = lane & 15;              // A row / B,D column within subtile
    const int colg = ni * 16 + lcol;         // column 0..63 inside this block

    // biases (scalar loads, L2-cached):  b1,b2 shape (2, NB, 64)
    const float bias1r = b1[(size_t)n * BS_ + colg];
    const float bias1i = b1[(size_t)(NB_ + n) * BS_ + colg];
    const float bias2r = b2[(size_t)n * BS_ + colg];
    const float bias2i = b2[(size_t)(NB_ + n) * BS_ + colg];

    // ---- layer 1: o1r = gelu(xr@w1r - xi@w1i + b1r), o1i = gelu(xi@w1r + xr@w1i + b1i)
    v8f accR, accI;
    #pragma unroll
    for (int v = 0; v < 8; ++v) { accR[v] = bias1r; accI[v] = bias1i; }

    accR = wmma_tile(Ar, Wr,  1.0f, accR, lcol, half, ni * 16);
    accR = wmma_tile(Ai, Wi, -1.0f, accR, lcol, half, ni * 16);
    accI = wmma_tile(Ai, Wr,  1.0f, accI, lcol, half, ni * 16);
    accI = wmma_tile(Ar, Wi,  1.0f, accI, lcol, half, ni * 16);

    // exact GELU, store hidden to LDS (D layout: VGPR v -> M = v + 8*half)
    #pragma unroll
    for (int v = 0; v < 8; ++v) {
        int row = v + 8 * half;
        float vr = accR[v];
        float vi = accI[v];
        vr = 0.5f * vr * (1.0f + erff(vr * 0.7071067811865476f));
        vi = 0.5f * vi * (1.0f + erff(vi * 0.7071067811865476f));
        Hr[row * 65 + colg] = vr;
        Hi[row * 65 + colg] = vi;
    }
    __syncthreads();   // layer-1 done everywhere; safe to overwrite weights

    // ---- stage layer-2 weights into the same LDS buffers
#if HAVE_TDM
    if (tid < 32) {
        tdm_load_2d(w2 + (size_t)n * 4096,         Wr, 4096, 1, 4096, false);
        tdm_load_2d(w2 + (size_t)(NB_ + n) * 4096, Wi, 4096, 1, 4096, false);
        __builtin_amdgcn_s_wait_tensorcnt(0);
    }
#else
    {
        const float4* w2r4 = (const float4*)(w2 + (size_t)n * 4096);
        const float4* w2i4 = (const float4*)(w2 + (size_t)(NB_ + n) * 4096);
        float4* Wr4 = (float4*)Wr;
        float4* Wi4 = (float4*)Wi;
        #pragma unroll
        for (int j = 0; j < 8; ++j) {
            int idx = tid + (j << 7);
            Wr4[idx] = w2r4[idx];
            Wi4[idx] = w2i4[idx];
        }
    }
#endif
    __syncthreads();

    // ---- layer 2: o2r = h_r@w2r - h_i@w2i + b2r ; o2i = h_i@w2r + h_r@w2i + b2i
    v8f oR, oI;
    #pragma unroll
    for (int v = 0; v < 8; ++v) { oR[v] = bias2r; oI[v] = bias2i; }

    oR = wmma_tile(Hr, Wr,  1.0f, oR, lcol, half, ni * 16);
    oR = wmma_tile(Hi, Wi, -1.0f, oR, lcol, half, ni * 16);
    oI = wmma_tile(Hi, Wr,  1.0f, oI, lcol, half, ni * 16);
    oI = wmma_tile(Hr, Wi,  1.0f, oI, lcol, half, ni * 16);

    // ---- write back in place (rows are exclusive to this workgroup)
    #pragma unroll
    for (int v = 0; v < 8; ++v) {
        int row = v + 8 * half;
        size_t goff = abase + (size_t)row * P_ + colg;
        gr[goff] = oR[v];
        gi[goff] = oI[v];
    }
}

// ---------------------------------------------------------------------------
// Kernel 3: Hermitian reconstruction + inverse FFT (ortho) + residual add.
// ---------------------------------------------------------------------------
__global__ __launch_bounds__(256)
void fft_inv_kernel(const float* __restrict__ gr, const float* __restrict__ gi,
                    const float* __restrict__ x, float* __restrict__ out)
{
    __shared__ float sre[2][2048];
    __shared__ float sim[2][2048];

    const int wg  = blockIdx.x;
    const int b   = wg >> 12;
    const int rem = wg & 4095;
    const int p   = rem >> 3;
    const int c   = rem & 7;
    const int tid = threadIdx.x;

    // load bins 0..1024 and mirror 1..1023 -> 2047..1025 (conjugate)
    #pragma unroll
    for (int i = 0; i < 5; ++i) {
        int l = tid + (i << 8);
        if (l < LF_) {
            size_t o = ((size_t)((b * LF_ + l) * C_ + c)) * P_ + p;
            float vr = gr[o];
            float vi = gi[o];
            sre[0][l] = vr;
            sim[0][l] = vi;
            if (l >= 1 && l <= 1023) {
                sre[0][2048 - l] =  vr;
                sim[0][2048 - l] = -vi;
            }
        }
    }
    __syncthreads();

    // inverse Stockham (sign = +1)
    int cur = 0;
    int logm = 0;
    #pragma unroll 1
    for (int stage = 0; stage < 11; ++stage) {
        const int m = 1 << logm;
        #pragma unroll
        for (int it = 0; it < 4; ++it) {
            const int t = tid + (it << 8);
            const int j = t >> logm;
            const int k = t & (m - 1);
            const float ang = 3.14159265358979323846f *
                              ((float)(j * m) * (1.0f / 1024.0f));
            const float wr = __cosf(ang);
            const float wi = __sinf(ang);
            const int i0 = k + (j << logm);
            const float ar = sre[cur][i0],        ai = sim[cur][i0];
            const float br = sre[cur][i0 + 1024], bi = sim[cur][i0 + 1024];
            const int o0 = k + (j << (logm + 1));
            sre[cur ^ 1][o0] = ar + br;
            sim[cur ^ 1][o0] = ai + bi;
            const float dr = ar - br, di = ai - bi;
            sre[cur ^ 1][o0 + m] = wr * dr - wi * di;
            sim[cur ^ 1][o0 + m] = wr * di + wi * dr;
        }
        __syncthreads();
        cur ^= 1;
        ++logm;
    }

    // real part * 1/sqrt(N) + residual
    const float* xbase = x + (size_t)b * (L_ * P_ * C_) + (size_t)p * C_ + c;
    float* obase = out + (size_t)b * (L_ * P_ * C_) + (size_t)p * C_ + c;
    #pragma unroll
    for (int i = 0; i < 8; ++i) {
        int l = tid + (i << 8);
        size_t off = (size_t)l * (P_ * C_);
        obase[off] = sre[cur][l] * ORTHO_SCALE + xbase[off];
    }
}

// ---------------------------------------------------------------------------
// Host launcher (graph-capture safe: kernel launches only).
// Workspace: spectrum real + imag, each B*LF*C*P floats (64.06 MiB each).
// ---------------------------------------------------------------------------
extern "C" void kernel_launch(void* const* d_in, const int* in_sizes, int n_in,
                              void* d_out, int out_size, void* d_ws, size_t ws_size,
                              hipStream_t stream)
{
    (void)in_sizes; (void)n_in; (void)out_size; (void)ws_size;

    const float* x  = (const float*)d_in[0];
    const float* w1 = (const float*)d_in[1];
    const float* b1 = (const float*)d_in[2];
    const float* w2 = (const float*)d_in[3];
    const float* b2 = (const float*)d_in[4];
    float* out = (float*)d_out;

    const size_t spec_elems = (size_t)B_ * LF_ * C_ * P_;   // 16,793,600
    float* wsR = (float*)d_ws;
    float* wsI = wsR + spec_elems;

    // 1) forward rfft: one WG per (b,p,c) signal
    fft_fwd_kernel<<<dim3(B_ * P_ * C_), dim3(256), 0, stream>>>(x, wsR, wsI);

    // 2) block MLP: (row-tiles, blocks); ROWS_ = 16 * 2050 exactly
    afno_mlp_kernel<<<dim3(ROWS_ / 16, NB_), dim3(128), 0, stream>>>(
        wsR, wsI, w1, b1, w2, b2);

    // 3) inverse rfft + residual
    fft_inv_kernel<<<dim3(B_ * P_ * C_), dim3(256), 0, stream>>>(wsR, wsI, x, out);
}